// eqAMPBC_31018253811802
// MI455X (gfx1250) — compile-verified
//
#include <hip/hip_runtime.h>
#include <stdint.h>

// ---------------------------------------------------------------------------
// CDNA5 (gfx1250) fused kernel for the FWM/XPM/IXIXPM equalizer reference.
// wave32, 1 batch element per lane; x tile staged in LDS via the gfx1250
// async global->LDS path (global_load_async_to_lds_b64 + s_wait_asynccnt).
// Flip USE_ASYNC_LDS to 0 for a plain-LDS fallback.
// ---------------------------------------------------------------------------
#ifndef USE_ASYNC_LDS
#define USE_ASYNC_LDS 1
#endif

#define BLK  128
#define MM   41
#define PH   20
#define KF   68
#define ROW  82              // floats per batch element per array ( [41][2] )
#define LDSH (BLK * ROW)     // floats per array region in LDS

// ---- compile-time FWM index tables (must match reference enumeration order)
struct FwmTables {
    unsigned u[KF][2];   // packed unit: a*2 | (b*2)<<8 | (c*2)<<16 | solo<<24
    unsigned nu[KF];     // units per k (1 or 2)
    int K;
};

constexpr unsigned fwm_pack(int a, int b, int c, int s) {
    return (unsigned)(2 * (a + PH)) |
           ((unsigned)(2 * (b + PH)) << 8) |
           ((unsigned)(2 * (c + PH)) << 16) |
           ((unsigned)s << 24);
}

constexpr FwmTables fwm_make() {
    FwmTables t{};
    int k = 0;
    for (int m = -PH; m <= PH; ++m) {
        for (int n = -PH; n <= PH; ++n) {
            int mn  = m * n;
            int amn = mn < 0 ? -mn : mn;
            int am  = m < 0 ? -m : m;
            int an  = n < 0 ? -n : n;
            if (!(amn <= PH && n >= am && am + an <= PH && mn != 0)) continue;
            if (n == m) {                     // pairs (m,m), (-m,-m): two solos
                t.u[k][0] = fwm_pack(m, m, 2 * m, 1);
                t.u[k][1] = fwm_pack(-m, -m, -2 * m, 1);
                t.nu[k] = 2;
            } else if (m + n == 0) {          // pairs (m,n),(n,m): one duo, c = center
                t.u[k][0] = fwm_pack(m, n, 0, 0);
                t.u[k][1] = fwm_pack(0, 0, 0, 1);  // unused
                t.nu[k] = 1;
            } else {                          // 4 pairs: two duos
                t.u[k][0] = fwm_pack(m, n, m + n, 0);
                t.u[k][1] = fwm_pack(-m, -n, -(m + n), 0);
                t.nu[k] = 2;
            }
            ++k;
        }
    }
    t.K = k;
    return t;
}

static_assert(fwm_make().K == KF, "FWM index count mismatch vs reference (HDIM)");

constexpr FwmTables g_fwm_host = fwm_make();
__constant__ FwmTables g_fwm = g_fwm_host;

// ---------------------------------------------------------------------------
__global__ __launch_bounds__(BLK) void eqampbc_kernel(
    const float* __restrict__ x_real, const float* __restrict__ x_imag,
    const float* __restrict__ task_info,
    const float* __restrict__ W_real, const float* __restrict__ W_imag,
    const float* __restrict__ conv1_w, const float* __restrict__ conv2_w,
    const float* __restrict__ C00p, float* __restrict__ out, int B)
{
    extern __shared__ float smem[];          // [LDSH] real | [LDSH] imag
    const int tid = threadIdx.x;
    const int b0  = blockIdx.x * BLK;
    const int b   = b0 + tid;
    const int nb  = (B - b0 < BLK) ? (B - b0) : BLK;
    const int npair = nb * (ROW / 2);        // float2 elements per array

    // ---- stage x tile into LDS (coalesced, contiguous region per block) ----
    {
        const float* gr = x_real + (size_t)b0 * ROW;
        const float* gi = x_imag + (size_t)b0 * ROW;
#if USE_ASYNC_LDS
        for (int p = tid; p < npair; p += BLK) {
            uint32_t l0 = (uint32_t)(p * 8);
            uint64_t a0 = (uint64_t)(uintptr_t)gr + (uint64_t)p * 8u;
            asm volatile("global_load_async_to_lds_b64 %0, %1, off"
                         :: "v"(l0), "v"(a0) : "memory");
            uint32_t l1 = l0 + (uint32_t)(LDSH * 4);
            uint64_t a1 = (uint64_t)(uintptr_t)gi + (uint64_t)p * 8u;
            asm volatile("global_load_async_to_lds_b64 %0, %1, off"
                         :: "v"(l1), "v"(a1) : "memory");
        }
        asm volatile("s_wait_asynccnt 0" ::: "memory");
#else
        const float2* gr2 = (const float2*)gr;
        const float2* gi2 = (const float2*)gi;
        float2* sr2 = (float2*)smem;
        float2* si2 = (float2*)(smem + LDSH);
        for (int p = tid; p < npair; p += BLK) { sr2[p] = gr2[p]; si2[p] = gi2[p]; }
#endif
    }
    __syncthreads();
    if (b >= B) return;

    const float* xr = smem + tid * ROW;          // [41][2] real (unscaled)
    const float* xi = smem + LDSH + tid * ROW;   // [41][2] imag (unscaled)

    // P = 10^(task/10)/2 ; the sqrt(P) input scaling + final 1/sqrt(P) fold to:
    //   out = P*(E(x)+ixi(x)) + x_c * exp(j*P*phi(x))
    const float P   = 0.5f * exp2f(task_info[(size_t)b * 4] * 0.3321928094887362f);
    const float c00 = C00p[0];

    // ---- FWM triplet features + K-contraction against Wc -------------------
    float E0r = 0.f, E0i = 0.f, E1r = 0.f, E1i = 0.f;

    auto term = [&](unsigned w, float& f0r, float& f0i, float& f1r, float& f1i) {
        const unsigned a    = w & 0xffu;
        const unsigned bb   = (w >> 8) & 0xffu;
        const unsigned c    = (w >> 16) & 0xffu;
        const unsigned solo = w >> 24;
        const float2 arv = *(const float2*)(xr + a);
        const float2 aiv = *(const float2*)(xi + a);
        const float2 crv = *(const float2*)(xr + c);
        const float2 civ = *(const float2*)(xi + c);
        // S_a = sum over modes of x_a * conj(x_c)   (A + A[::-1] is mode-shared)
        const float Sar = arv.x * crv.x + aiv.x * civ.x + arv.y * crv.y + aiv.y * civ.y;
        const float Sai = aiv.x * crv.x - arv.x * civ.x + aiv.y * crv.y - arv.y * civ.y;
        if (solo) {                      // single term: feats += S_a * x_a
            f0r += Sar * arv.x - Sai * aiv.x;
            f0i += Sar * aiv.x + Sai * arv.x;
            f1r += Sar * arv.y - Sai * aiv.y;
            f1i += Sar * aiv.y + Sai * arv.y;
        } else {                         // duo: feats += S_a*x_b + S_b*x_a
            const float2 brv = *(const float2*)(xr + bb);
            const float2 biv = *(const float2*)(xi + bb);
            const float Sbr = brv.x * crv.x + biv.x * civ.x + brv.y * crv.y + biv.y * civ.y;
            const float Sbi = biv.x * crv.x - brv.x * civ.x + biv.y * crv.y - brv.y * civ.y;
            f0r += Sar * brv.x - Sai * biv.x + Sbr * arv.x - Sbi * aiv.x;
            f0i += Sar * biv.x + Sai * brv.x + Sbr * aiv.x + Sbi * arv.x;
            f1r += Sar * brv.y - Sai * biv.y + Sbr * arv.y - Sbi * aiv.y;
            f1i += Sar * biv.y + Sai * brv.y + Sbr * aiv.y + Sbi * arv.y;
        }
    };

#pragma unroll 2
    for (int k = 0; k < KF; ++k) {
        float f0r = 0.f, f0i = 0.f, f1r = 0.f, f1i = 0.f;
        term(g_fwm.u[k][0], f0r, f0i, f1r, f1i);
        if (g_fwm.nu[k] == 2) term(g_fwm.u[k][1], f0r, f0i, f1r, f1i);
        // E_i += feats_i * (W_real + j W_imag)[i,k]   (uniform -> scalar loads)
        const float wr0 = W_real[k],      wi0 = W_imag[k];
        const float wr1 = W_real[KF + k], wi1 = W_imag[KF + k];
        E0r += f0r * wr0 - f0i * wi0;  E0i += f0r * wi0 + f0i * wr0;
        E1r += f1r * wr1 - f1i * wi1;  E1i += f1r * wi1 + f1i * wr1;
    }

    // ---- XPM phase + IXIXPM: taps split around the zeroed center -----------
    float acc0 = 0.f, acc1 = 0.f, fr = 0.f, fi = 0.f;

    auto tap = [&](int p) {
        const float2 rv = *(const float2*)(xr + 2 * p);
        const float2 iv = *(const float2*)(xi + 2 * p);
        const float pw0 = rv.x * rv.x + iv.x * iv.x;
        const float pw1 = rv.y * rv.y + iv.y * iv.y;
        const float w1 = conv1_w[p], w2 = conv2_w[p];
        acc0 += w1 * (2.f * pw0 + pw1);
        acc1 += w1 * (2.f * pw1 + pw0);
        fr   += w2 * (rv.x * rv.y + iv.x * iv.y);   // f0 = sum u[p][0]*w2
        fi   += w2 * (iv.x * rv.y - rv.x * iv.y);   // (f1 = conj(f0))
    };

    for (int p = 0; p < PH; ++p)      tap(p);
    for (int p = PH + 1; p < MM; ++p) tap(p);

    const float2 rc = *(const float2*)(xr + 2 * PH);
    const float2 ic = *(const float2*)(xi + 2 * PH);
    const float r0c = rc.x, i0c = ic.x, r1c = rc.y, i1c = ic.y;
    const float pwc = r0c * r0c + i0c * i0c + r1c * r1c + i1c * i1c;

    const float phi0 = P * (c00 * pwc + 2.f * acc0);
    const float phi1 = P * (c00 * pwc + 2.f * acc1);
    float s0, c0, s1, c1;
    __sincosf(phi0, &s0, &c0);
    __sincosf(phi1, &s1, &c1);

    // ixi0 = j*xc1*f0 ; ixi1 = j*xc0*conj(f0)
    const float ixi0r = -(r1c * fi + i1c * fr);
    const float ixi0i =   r1c * fr - i1c * fi;
    const float ixi1r =   r0c * fi - i0c * fr;
    const float ixi1i =   r0c * fr + i0c * fi;

    float4 o;
    o.x = P * (E0r + ixi0r) + (r0c * c0 - i0c * s0);
    o.y = P * (E0i + ixi0i) + (r0c * s0 + i0c * c0);
    o.z = P * (E1r + ixi1r) + (r1c * c1 - i1c * s1);
    o.w = P * (E1i + ixi1i) + (r1c * s1 + i1c * c1);
    *(float4*)(out + (size_t)b * 4) = o;
}

// ---------------------------------------------------------------------------
extern "C" void kernel_launch(void* const* d_in, const int* in_sizes, int n_in,
                              void* d_out, int out_size, void* d_ws, size_t ws_size,
                              hipStream_t stream) {
    (void)n_in; (void)out_size; (void)d_ws; (void)ws_size;
    const float* x_real    = (const float*)d_in[0];
    const float* x_imag    = (const float*)d_in[1];
    const float* task_info = (const float*)d_in[2];
    const float* W_real    = (const float*)d_in[3];
    const float* W_imag    = (const float*)d_in[4];
    const float* conv1_w   = (const float*)d_in[5];
    const float* conv2_w   = (const float*)d_in[6];
    const float* C00       = (const float*)d_in[7];
    float* out = (float*)d_out;

    const int B    = in_sizes[0] / ROW;            // x_real is [B,41,2]
    const int grid = (B + BLK - 1) / BLK;
    const size_t smem = (size_t)2 * LDSH * sizeof(float);   // 84 KB of 320 KB LDS
                                                            // -> 3 blocks / WGP

    eqampbc_kernel<<<grid, BLK, smem, stream>>>(
        x_real, x_imag, task_info, W_real, W_imag, conv1_w, conv2_w, C00, out, B);
}